// SelfAttention_5617817223831
// MI455X (gfx1250) — compile-verified
//
#include <hip/hip_runtime.h>
#include <hip/hip_bf16.h>
#include <math.h>

// ---------------------------------------------------------------------------
// Problem constants (from reference): B=32, S=1, DIM=4096, HQ=32, HKV=8,
// HD=128, MAX_SEQ=2048, START_POS=2047 -> t = 2048 attended positions.
// ---------------------------------------------------------------------------
#define BATCH    32
#define DIM      4096
#define HQ       32
#define HKV      8
#define HD       128
#define NREP     4            // HQ / HKV
#define TSEQ     2048         // START_POS + S
#define KVROW    (HKV * HD)   // 1024 floats per (b, t) in cache
#define KTILE    32           // K rows staged per async tile (16 KB)

typedef __attribute__((ext_vector_type(2))) float v2f;
typedef __attribute__((ext_vector_type(8))) float v8f;

// ---- gfx1250 async-to-LDS path (guarded; falls back to sync copy) ---------
#if defined(__has_builtin)
#  if __has_builtin(__builtin_amdgcn_global_load_async_to_lds_b128)
#    define HAVE_ASYNC_LDS 1
#  endif
#endif
#ifndef HAVE_ASYNC_LDS
#  define HAVE_ASYNC_LDS 0
#endif

#if HAVE_ASYNC_LDS
// Probe-confirmed (round-2 diagnostic): param 0 is `int __vector(4) __device__*`
// i.e. v4i in addrspace(1); dst is the LDS-side v4i pointer; then imm offset, cpol.
typedef int v4i __attribute__((vector_size(16)));
typedef __attribute__((address_space(1))) v4i gv4i;   // global-AS 16B vector
typedef __attribute__((address_space(3))) v4i lv4i;   // LDS-AS 16B vector
#  if __has_builtin(__builtin_amdgcn_s_wait_asynccnt)
#    define WAIT_ASYNC(n) __builtin_amdgcn_s_wait_asynccnt(n)
#  else
#    define WAIT_ASYNC(n) asm volatile("s_wait_asynccnt %0" :: "i"(n) : "memory")
#  endif
#else
#  define WAIT_ASYNC(n) do { } while (0)
#endif

// ---------------------------------------------------------------------------
// GEMM: C[m][n] = sum_k A[m][k] * W[n][k]   (A: MxK row-major, W: NxK row-major)
// One wave computes one 16x16 tile of C via V_WMMA_F32_16X16X4_F32.
//
// WMMA f32 16x16x4 VGPR layouts (ISA 7.12.2):
//   A (16x4):  lanes 0-15 hold M=lane%16; VGPR0 = K=0 (+2 for lanes 16-31),
//              VGPR1 = K=1 (+2 for lanes 16-31)   -> per-lane float2 at k+2*half
//   B (4x16):  lanes 0-15 hold N=lane%16; same K split across lane halves
//   C/D:       VGPR j: lanes 0-15 -> M=j, lanes 16-31 -> M=j+8, N=lane%16
// ---------------------------------------------------------------------------
__global__ __launch_bounds__(128) void gemm_xwT_wmma(
    const float* __restrict__ A, const float* __restrict__ W,
    float* __restrict__ C, int M, int N, int K)
{
    const int wavesPerBlock = blockDim.x >> 5;
    const int wave = blockIdx.x * wavesPerBlock + (threadIdx.x >> 5);
    const int lane = threadIdx.x & 31;
    const int tilesM = M >> 4;
    const int tilesN = N >> 4;
    const int tm = wave % tilesM;        // uniform within a wave
    const int tn = wave / tilesM;
    if (tn >= tilesN) return;            // wave-uniform branch: EXEC stays full

    const int half = lane >> 4;          // selects K pair (0,1) vs (2,3)
    const int l16  = lane & 15;

    const float* __restrict__ arow = A + (size_t)(tm * 16 + l16) * K;
    const float* __restrict__ wrow = W + (size_t)(tn * 16 + l16) * K;

    v8f acc = {};
    #pragma unroll 8
    for (int k = 0; k < K; k += 4) {
        v2f a = *(const v2f*)(arow + k + 2 * half);
        v2f b = *(const v2f*)(wrow + k + 2 * half);
        // 8 args: (neg_a, A, neg_b, B, c_mod, C, reuse_a, reuse_b)
        acc = __builtin_amdgcn_wmma_f32_16x16x4_f32(
            false, a, false, b, (short)0, acc, false, false);
    }

    float* __restrict__ crow = C + (size_t)(tm * 16 + 8 * half) * N + tn * 16 + l16;
    #pragma unroll
    for (int j = 0; j < 8; ++j)
        crow[(size_t)j * N] = acc[j];
}

// ---------------------------------------------------------------------------
// RoPE, in place, interleaved-pair convention:
//   (e0, e1) -> (e0*c - e1*s, e0*s + e1*c), pair p = (index%128)/2
// ---------------------------------------------------------------------------
#define QPAIRS (BATCH * HQ * (HD / 2))   // 65536
#define KPAIRS (BATCH * HKV * (HD / 2))  // 16384

__global__ __launch_bounds__(256) void rope_inplace(
    float* __restrict__ qbuf, float* __restrict__ kbuf,
    const float* __restrict__ fc, const float* __restrict__ fs)
{
    int idx = blockIdx.x * blockDim.x + threadIdx.x;
    if (idx >= QPAIRS + KPAIRS) return;
    float* buf;
    int p;
    if (idx < QPAIRS) {
        buf = qbuf + 2 * (size_t)idx;
        p = idx & 63;
    } else {
        int i2 = idx - QPAIRS;
        buf = kbuf + 2 * (size_t)i2;
        p = i2 & 63;
    }
    const float c = fc[p], s = fs[p];
    const float t0 = buf[0], t1 = buf[1];
    buf[0] = t0 * c - t1 * s;
    buf[1] = t0 * s + t1 * c;
}

// ---------------------------------------------------------------------------
// Attention: one workgroup (256 threads) per (b, g), g = kv head; 4 q heads
// share the kv head. K rows are double-buffered into LDS with
// GLOBAL_LOAD_ASYNC_TO_LDS_B128 (ASYNCcnt-paced), dots are computed from LDS
// (8 lanes per row, __shfl_xor wave32 reduction), then two-pass softmax and a
// coalesced, prefetch-assisted V-weighted sum. Position TSEQ-1 reads the
// freshly-RoPE'd k / new v from workspace so input caches are never mutated.
// ---------------------------------------------------------------------------
__device__ __forceinline__ void stage_k_tile(
    const float* __restrict__ cache_k, const float* __restrict__ knew,
    int b, int g, int tid, int t0, float (*dstTile)[HD])
{
    // KTILE rows * 512B = 16KB; 256 threads * 16B = 4KB per round; 4 rounds.
    #pragma unroll
    for (int rnd = 0; rnd < 4; ++rnd) {
        int c   = tid + rnd * 256;       // 16B chunk index (512/16 = 32 per row)
        int row = c >> 5;                // 0..KTILE-1
        int col = (c & 31) * 4;          // float index within row
        int t   = t0 + row;
        const float* src = (t < TSEQ - 1)
            ? cache_k + (((size_t)b * TSEQ + t) * HKV + g) * HD + col
            : knew + ((size_t)b * HKV + g) * HD + col;
        float* dst = &dstTile[row][col];
#if HAVE_ASYNC_LDS
        __builtin_amdgcn_global_load_async_to_lds_b128(
            (gv4i*)src, (lv4i*)dst, 0, 0);
#else
        *(float4*)dst = *(const float4*)src;
#endif
    }
}

__global__ __launch_bounds__(256) void attn_decode(
    const float* __restrict__ qbuf,      // (B, HQ, HD)
    const float* __restrict__ knew,      // (B, HKV, HD)
    const float* __restrict__ vnew,      // (B, HKV, HD)
    const float* __restrict__ cache_k,   // (B, MAX_SEQ, HKV, HD)
    const float* __restrict__ cache_v,
    float* __restrict__ obuf)            // (B, HQ, HD)
{
    const int b = blockIdx.x >> 3;
    const int g = blockIdx.x & 7;
    const int tid = threadIdx.x;

    __shared__ float sQ[NREP * HD];          //  2 KB
    __shared__ float sS[NREP][TSEQ];         // 32 KB
    __shared__ float sRed[NREP][256];        //  4 KB
    __shared__ float sK[2][KTILE][HD];       // 32 KB double buffer

    // Load the 4 query heads for this kv group.
    for (int i = tid; i < NREP * HD; i += 256)
        sQ[i] = qbuf[((size_t)b * HQ + g * NREP) * HD + i];
    __syncthreads();

    const float scale = 0.08838834764831844f;  // 1/sqrt(128)

    // ---- Pass 1: scores via async-staged K tiles -----------------------
    const int rw = tid >> 3;            // row within tile: 0..31
    const int dc = (tid & 7) * 16;      // 16-dim chunk of HD handled by lane
    const bool writer = (tid & 7) == 0;

    float lmax[NREP] = {-INFINITY, -INFINITY, -INFINITY, -INFINITY};

    stage_k_tile(cache_k, knew, b, g, tid, 0, sK[0]);      // prime buffer 0

    const int NTILE = TSEQ / KTILE;     // 64
    for (int ti = 0; ti < NTILE; ++ti) {
        const int buf = ti & 1;
        if (ti + 1 < NTILE)
            stage_k_tile(cache_k, knew, b, g, tid, (ti + 1) * KTILE, sK[buf ^ 1]);
        // 4 ops outstanding for the next tile; wait until this tile's 4 landed.
        if (ti + 1 < NTILE) { WAIT_ASYNC(4); } else { WAIT_ASYNC(0); }
        __syncthreads();

        float p0 = 0.f, p1 = 0.f, p2 = 0.f, p3 = 0.f;
        const float* krow = &sK[buf][rw][dc];
        #pragma unroll
        for (int dd = 0; dd < 16; dd += 4) {
            float4 kv = *(const float4*)(krow + dd);
            p0 += sQ[0 * HD + dc + dd] * kv.x + sQ[0 * HD + dc + dd + 1] * kv.y
                + sQ[0 * HD + dc + dd + 2] * kv.z + sQ[0 * HD + dc + dd + 3] * kv.w;
            p1 += sQ[1 * HD + dc + dd] * kv.x + sQ[1 * HD + dc + dd + 1] * kv.y
                + sQ[1 * HD + dc + dd + 2] * kv.z + sQ[1 * HD + dc + dd + 3] * kv.w;
            p2 += sQ[2 * HD + dc + dd] * kv.x + sQ[2 * HD + dc + dd + 1] * kv.y
                + sQ[2 * HD + dc + dd + 2] * kv.z + sQ[2 * HD + dc + dd + 3] * kv.w;
            p3 += sQ[3 * HD + dc + dd] * kv.x + sQ[3 * HD + dc + dd + 1] * kv.y
                + sQ[3 * HD + dc + dd + 2] * kv.z + sQ[3 * HD + dc + dd + 3] * kv.w;
        }
        // Reduce across the 8 lanes sharing this row (same wave32).
        #pragma unroll
        for (int m = 1; m < 8; m <<= 1) {
            p0 += __shfl_xor(p0, m);
            p1 += __shfl_xor(p1, m);
            p2 += __shfl_xor(p2, m);
            p3 += __shfl_xor(p3, m);
        }
        if (writer) {
            const int t = ti * KTILE + rw;
            p0 *= scale; p1 *= scale; p2 *= scale; p3 *= scale;
            sS[0][t] = p0; sS[1][t] = p1; sS[2][t] = p2; sS[3][t] = p3;
            lmax[0] = fmaxf(lmax[0], p0); lmax[1] = fmaxf(lmax[1], p1);
            lmax[2] = fmaxf(lmax[2], p2); lmax[3] = fmaxf(lmax[3], p3);
        }
        __syncthreads();   // everyone done with sK[buf] before it is re-staged
    }

    // ---- Max reduction -------------------------------------------------
    #pragma unroll
    for (int r = 0; r < NREP; ++r) sRed[r][tid] = lmax[r];
    __syncthreads();
    for (int s = 128; s > 0; s >>= 1) {
        if (tid < s) {
            #pragma unroll
            for (int r = 0; r < NREP; ++r)
                sRed[r][tid] = fmaxf(sRed[r][tid], sRed[r][tid + s]);
        }
        __syncthreads();
    }
    float gmax[NREP];
    #pragma unroll
    for (int r = 0; r < NREP; ++r) gmax[r] = sRed[r][0];
    __syncthreads();

    // ---- exp + sum -----------------------------------------------------
    float lsum[NREP] = {0.f, 0.f, 0.f, 0.f};
    for (int t = tid; t < TSEQ; t += 256) {
        #pragma unroll
        for (int r = 0; r < NREP; ++r) {
            float p = __expf(sS[r][t] - gmax[r]);
            sS[r][t] = p;
            lsum[r] += p;
        }
    }
    #pragma unroll
    for (int r = 0; r < NREP; ++r) sRed[r][tid] = lsum[r];
    __syncthreads();
    for (int s = 128; s > 0; s >>= 1) {
        if (tid < s) {
            #pragma unroll
            for (int r = 0; r < NREP; ++r)
                sRed[r][tid] += sRed[r][tid + s];
        }
        __syncthreads();
    }
    const int r0 = tid >> 7;        // 0 or 1
    const int r1 = r0 + 2;
    const int d  = tid & (HD - 1);
    const float inv0 = 1.0f / sRed[r0][0];
    const float inv1 = 1.0f / sRed[r1][0];
    __syncthreads();

    // ---- V-weighted sum (lanes cover contiguous d -> coalesced) --------
    float acc0 = 0.f, acc1 = 0.f;
    const float* vbase = cache_v + ((size_t)b * TSEQ * HKV + g) * HD + d;
    for (int t = 0; t < TSEQ - 1; ++t) {
        // stream-prefetch ~48 rows (192KB of this block's V stream) ahead
        int tp = (t + 48 < TSEQ - 1) ? t + 48 : t;
        __builtin_prefetch(vbase + (size_t)tp * KVROW, 0, 0);
        float v = vbase[(size_t)t * KVROW];
        acc0 += sS[r0][t] * v;
        acc1 += sS[r1][t] * v;
    }
    {   // last position: the new v lives in workspace
        float v = vnew[((size_t)b * HKV + g) * HD + d];
        acc0 += sS[r0][TSEQ - 1] * v;
        acc1 += sS[r1][TSEQ - 1] * v;
    }
    obuf[((size_t)b * HQ + g * NREP + r0) * HD + d] = acc0 * inv0;
    obuf[((size_t)b * HQ + g * NREP + r1) * HD + d] = acc1 * inv1;
}

// ---------------------------------------------------------------------------
// Launch
// ---------------------------------------------------------------------------
extern "C" void kernel_launch(void* const* d_in, const int* in_sizes, int n_in,
                              void* d_out, int out_size, void* d_ws, size_t ws_size,
                              hipStream_t stream)
{
    const float* x       = (const float*)d_in[0];  // (32, 1, 4096)
    const float* wq      = (const float*)d_in[1];  // (4096, 4096) n-major
    const float* wk      = (const float*)d_in[2];  // (1024, 4096)
    const float* wv      = (const float*)d_in[3];  // (1024, 4096)
    const float* wo      = (const float*)d_in[4];  // (4096, 4096)
    const float* cache_k = (const float*)d_in[5];  // (32, 2048, 8, 128)
    const float* cache_v = (const float*)d_in[6];
    const float* fc      = (const float*)d_in[7];  // (1, 64)
    const float* fs      = (const float*)d_in[8];
    float* out = (float*)d_out;                    // (32, 4096)

    // Workspace carve-up (floats): q 131072 | k 32768 | v 32768 | attn 131072
    float* q_buf = (float*)d_ws;
    float* k_buf = q_buf + (size_t)BATCH * HQ * HD;
    float* v_buf = k_buf + (size_t)BATCH * HKV * HD;
    float* a_buf = v_buf + (size_t)BATCH * HKV * HD;

    // QKV projections (WMMA GEMMs). tiles = (M/16)*(N/16); 4 waves / block.
    {
        dim3 blk(128);
        int tilesQ = (BATCH / 16) * (HQ * HD / 16);   // 512 waves
        int tilesK = (BATCH / 16) * (HKV * HD / 16);  // 128 waves
        gemm_xwT_wmma<<<dim3(tilesQ / 4), blk, 0, stream>>>(x, wq, q_buf, BATCH, HQ * HD, DIM);
        gemm_xwT_wmma<<<dim3(tilesK / 4), blk, 0, stream>>>(x, wk, k_buf, BATCH, HKV * HD, DIM);
        gemm_xwT_wmma<<<dim3(tilesK / 4), blk, 0, stream>>>(x, wv, v_buf, BATCH, HKV * HD, DIM);
    }

    // RoPE on q and k (in place).
    {
        int pairs = QPAIRS + KPAIRS;               // 81920
        rope_inplace<<<dim3((pairs + 255) / 256), dim3(256), 0, stream>>>(q_buf, k_buf, fc, fs);
    }

    // Attention: one workgroup per (b, kv-head).
    attn_decode<<<dim3(BATCH * HKV), dim3(256), 0, stream>>>(
        q_buf, k_buf, v_buf, cache_k, cache_v, a_buf);

    // Output projection directly into d_out.
    {
        int tiles = (BATCH / 16) * (DIM / 16);     // 512 waves
        gemm_xwT_wmma<<<dim3(tiles / 4), dim3(128), 0, stream>>>(a_buf, wo, out, BATCH, DIM, HQ * HD);
    }
}